// LookupFreeQuantizer_42202348650760
// MI455X (gfx1250) — compile-verified
//
#include <hip/hip_runtime.h>

// LFQ (lookup-free quantizer) forward for MI455X (gfx1250, wave32).
// GEMM [65536x10]x[10x1024] done with V_WMMA_F32_16X16X4_F32 (K padded 10->12,
// 3 WMMAs per 16x16 tile). Softmax row-max computed analytically as
// 200*sum|z| (codebook covers all sign patterns). Codebook values are
// regenerated from bits (identical to _make_codebook), so B costs no memory.

typedef __attribute__((ext_vector_type(2))) float v2f;
typedef __attribute__((ext_vector_type(8))) float v8f;

#define CH        10
#define INV_TEMP  200.0f      // 2 / ENT_TEMP
#define S_TOTAL   65536       // 64*32*32
#define Z_TOTAL   655360      // 64*10*32*32

// ws layout (floats): [0..1023] avg_probs accumulator, [1024] sum H, [1025] sum commit
__global__ __launch_bounds__(256) void lfq_main(
    const float* __restrict__ z, float* __restrict__ zq,
    float* __restrict__ idx_out, float* __restrict__ gP,
    float* __restrict__ gH, float* __restrict__ gC)
{
  __shared__ float m_lds[16];     // 200*sum|z| per sample row
  __shared__ float Zl[16 * 8];    // per-wave partial Z per row
  __shared__ float Tl[16 * 8];    // per-wave partial sum(e*u) per row
  __shared__ float Zf[16];        // final Z per row

  const int tid = threadIdx.x;
  const int s0  = blockIdx.x * 16;     // 4096 blocks * 16 samples = 65536

  // Phase 0: per-sample scalars, sign output, index pack (16 lanes)
  if (tid < 16) {
    const int s  = s0 + tid;
    const int bb = s >> 10, hw = s & 1023;
    const float* zb = z  + bb * 10240 + hw;
    float*       qb = zq + bb * 10240 + hw;
    float sa = 0.f, com = 0.f; int idx = 0;
#pragma unroll
    for (int c = 0; c < CH; ++c) {
      float v  = zb[c * 1024];
      float av = fabsf(v);
      sa += av;
      float d = 1.f - av;            // (sign(v)-v)^2 == (1-|v|)^2
      com += d * d;
      if (v > 0.f) idx |= (1 << c);
      qb[c * 1024] = (v > 0.f) ? 1.f : -1.f;
    }
    m_lds[tid] = INV_TEMP * sa;      // analytic row max of affinity
    idx_out[s] = (float)idx;
    atomicAdd(gC, com);
  }
  __syncthreads();

  const int lane = tid & 31;
  const int wave = tid >> 5;          // 8 waves, each owns 128 codes
  const int half = lane >> 4;         // 0: rows 0..7 / K 0,1 ; 1: rows 8..15 / K 2,3
  const int ml   = lane & 15;

  float mrow[8];
#pragma unroll
  for (int r = 0; r < 8; ++r) mrow[r] = m_lds[r + 8 * half];

  // A tile: 16 samples x 12 channels (f32 WMMA layout, 3 K-chunks of 4)
  v2f A[3];
  {
    const int s  = s0 + ml;
    const int bb = s >> 10, hw = s & 1023;
    const float* zb = z + bb * 10240 + hw;
#pragma unroll
    for (int kc = 0; kc < 3; ++kc) {
      const int c0 = kc * 4 + half * 2;
      A[kc][0] = (c0     < CH) ? zb[c0 * 1024]       : 0.f;
      A[kc][1] = (c0 + 1 < CH) ? zb[(c0 + 1) * 1024] : 0.f;
    }
  }

  float Zp[8], Tp[8], es[64];
#pragma unroll
  for (int r = 0; r < 8; ++r) { Zp[r] = 0.f; Tp[r] = 0.f; }

#pragma unroll
  for (int t = 0; t < 8; ++t) {
    const int n = wave * 128 + t * 16 + ml;   // code index for this B column
    v2f B[3];
#pragma unroll
    for (int kc = 0; kc < 3; ++kc) {
      const int c0 = kc * 4 + half * 2;
      B[kc][0] = (c0     < CH) ? (((n >> c0)       & 1) ? 1.f : -1.f) : 0.f;
      B[kc][1] = (c0 + 1 < CH) ? (((n >> (c0 + 1)) & 1) ? 1.f : -1.f) : 0.f;
    }
    v8f C = {0.f, 0.f, 0.f, 0.f, 0.f, 0.f, 0.f, 0.f};
    C = __builtin_amdgcn_wmma_f32_16x16x4_f32(false, A[0], false, B[0], (short)0, C, false, false);
    C = __builtin_amdgcn_wmma_f32_16x16x4_f32(false, A[1], false, B[1], (short)0, C, false, false);
    C = __builtin_amdgcn_wmma_f32_16x16x4_f32(false, A[2], false, B[2], (short)0, C, false, false);
#pragma unroll
    for (int r = 0; r < 8; ++r) {
      float u = C[r] * INV_TEMP - mrow[r];    // <= ~0 : safe exp
      float e = __expf(u);
      es[t * 8 + r] = e;
      Zp[r] += e;
      Tp[r] += e * u;                          // for H = logZ - T/Z
    }
  }

  // reduce across the 16 lanes of each half (butterfly keeps halves separate)
#pragma unroll
  for (int r = 0; r < 8; ++r) {
#pragma unroll
    for (int m = 1; m < 16; m <<= 1) {
      Zp[r] += __shfl_xor(Zp[r], m, 32);
      Tp[r] += __shfl_xor(Tp[r], m, 32);
    }
  }
  if (ml == 0) {
#pragma unroll
    for (int r = 0; r < 8; ++r) {
      const int row = r + 8 * half;
      Zl[row * 8 + wave] = Zp[r];
      Tl[row * 8 + wave] = Tp[r];
    }
  }
  __syncthreads();
  if (tid < 16) {
    float zf = 0.f, tf = 0.f;
#pragma unroll
    for (int w = 0; w < 8; ++w) { zf += Zl[tid * 8 + w]; tf += Tl[tid * 8 + w]; }
    Zf[tid] = zf;
    atomicAdd(gH, __logf(zf) - tf / zf);       // per-sample entropy contribution
  }
  __syncthreads();

  float zr[8];
#pragma unroll
  for (int r = 0; r < 8; ++r) zr[r] = 1.f / Zf[r + 8 * half];

  // avg_probs: sum probs over the block's 16 rows per code, one atomic per code
#pragma unroll
  for (int t = 0; t < 8; ++t) {
    float cs = 0.f;
#pragma unroll
    for (int r = 0; r < 8; ++r) cs += es[t * 8 + r] * zr[r];
    cs += __shfl_xor(cs, 16, 32);              // combine the two M-halves (same N)
    if (half == 0)
      atomicAdd(gP + wave * 128 + t * 16 + ml, cs * (1.0f / (float)S_TOTAL));
  }
}

__global__ __launch_bounds__(1024) void lfq_finalize(const float* __restrict__ ws,
                                                     float* __restrict__ out5)
{
  __shared__ float red[1024];
  const int t = threadIdx.x;
  float p = ws[t];
  red[t] = -p * __logf(p + 1e-5f);
  __syncthreads();
  for (int s = 512; s > 0; s >>= 1) {
    if (t < s) red[t] += red[t + s];
    __syncthreads();
  }
  if (t == 0) {
    float avg_entropy = red[0];                          // ENT_GAMMA = 1
    float pse         = ws[1024] * (1.0f / (float)S_TOTAL);
    float commit      = 0.25f * ws[1025] * (1.0f / (float)Z_TOTAL);
    float ent         = 0.1f * (pse - avg_entropy);
    out5[0] = commit + ent;   // loss
    out5[1] = commit;         // commitment_loss
    out5[2] = ent;            // entropy_loss
    out5[3] = pse;            // per_sample_entropy
    out5[4] = avg_entropy;    // avg_entropy
  }
}

extern "C" void kernel_launch(void* const* d_in, const int* in_sizes, int n_in,
                              void* d_out, int out_size, void* d_ws, size_t ws_size,
                              hipStream_t stream) {
  const float* z = (const float*)d_in[0];
  // d_in[1] (codebook) is regenerated on the fly from bit patterns (bit-exact
  // with _make_codebook), so it is not read.
  float* out  = (float*)d_out;
  float* zq   = out;                 // [655360] z_q (b,c,h,w)
  float* scal = out + Z_TOTAL;       // [5] loss, commit, ent, pse, avg_ent
  float* idx  = out + Z_TOTAL + 5;   // [65536] indices (as float)
  float* ws   = (float*)d_ws;        // [1024] avg_probs acc, [1] H sum, [1] commit sum

  hipMemsetAsync(d_ws, 0, 1026 * sizeof(float), stream);
  lfq_main<<<S_TOTAL / 16, 256, 0, stream>>>(z, zq, idx, ws, ws + 1024, ws + 1025);
  lfq_finalize<<<1, 1024, 0, stream>>>(ws, scal);
}